// ProbabilityEstimatorCovarianceDiagonalizingKernel_11295763989015
// MI455X (gfx1250) — compile-verified
//
#include <hip/hip_runtime.h>
#include <math.h>

typedef __attribute__((ext_vector_type(2))) float v2f;
typedef __attribute__((ext_vector_type(8))) float v8f;

#define TDIM 512
#define KDIM 8
#define DDIM 8
#define EPSV 1e-7f
#define GAINV 3.5f
#define MAXLV 3.5f
#define TWO_PI_F 6.283185307179586f
#define LOG2E_F 1.4426950408889634f

// ---------------------------------------------------------------------------
// Workspace layout (floats):
//   aq   : [T,K, 2 chunks, 32 lanes, 2]  pre-swizzled A-fragments of Q^T
//   ct   : [T,K,8]   Q^T @ center       (folded center rotation)
//   ivl  : [T,K,8]   (0.5/var)*log2(e)  (base-2 pre-scaled)
//   lnc2 : [T,K,8]   -0.5*log2(2*pi*var)
//   w    : [T,K]     softmax weights
// ---------------------------------------------------------------------------

// ---------------------------------------------------------------------------
// Kernel 1: per-(t,k) Cayley transform Q = (I - A)(I + A)^-1 (8x8); emit the
// A-fragment table for V_WMMA_F32_16X16X4_F32 (lane L<16 -> M=L, K={0,1};
// lanes 16-31 -> K={2,3}; M rows 8..15 zero-padded), rotated centers, and
// base-2 variance terms. 4096 threads -> negligible runtime.
// ---------------------------------------------------------------------------
__global__ __launch_bounds__(256) void cayley_setup_kernel(
    const float* __restrict__ cov_param,     // [T,K,28]
    const float* __restrict__ log_variance,  // [T,K,8]
    const float* __restrict__ centers,       // [T,K,8]
    float* __restrict__ aq,                  // [T,K,2,32,2]
    float* __restrict__ ct,                  // [T,K,8]
    float* __restrict__ ivl,                 // [T,K,8]
    float* __restrict__ lnc2) {              // [T,K,8]
  const int tk = blockIdx.x * blockDim.x + threadIdx.x;
  if (tk >= TDIM * KDIM) return;

  const float* v = cov_param + tk * (DDIM * (DDIM - 1) / 2);  // 28 values

  // ut = reshape(concat(v, reverse(v), zeros(8)), 8,8) masked strict-upper
  float ut[8][8];
#pragma unroll
  for (int i = 0; i < 8; i++) {
#pragma unroll
    for (int j = 0; j < 8; j++) {
      const int f = i * 8 + j;
      float val = 0.f;
      if (f < 28)      val = v[f];
      else if (f < 56) val = v[55 - f];
      ut[i][j] = (j > i) ? val : 0.f;
    }
  }
  // antisymmetric A = 0.5*(ut^T - ut)
  float A[8][8];
#pragma unroll
  for (int i = 0; i < 8; i++)
#pragma unroll
    for (int j = 0; j < 8; j++) A[i][j] = 0.5f * (ut[j][i] - ut[i][j]);

  // Invert M = I + A (Gauss-Jordan; diag==1, antisym -> well conditioned)
  float M[8][8], Inv[8][8];
  for (int i = 0; i < 8; i++)
    for (int j = 0; j < 8; j++) {
      M[i][j]   = ((i == j) ? 1.f : 0.f) + A[i][j];
      Inv[i][j] = (i == j) ? 1.f : 0.f;
    }
  for (int col = 0; col < 8; col++) {
    const float r = 1.f / M[col][col];
    for (int j = 0; j < 8; j++) { M[col][j] *= r; Inv[col][j] *= r; }
    for (int row = 0; row < 8; row++) {
      if (row == col) continue;
      const float f = M[row][col];
      for (int j = 0; j < 8; j++) {
        M[row][j]   -= f * M[col][j];
        Inv[row][j] -= f * Inv[col][j];
      }
    }
  }

  // q[i][j] = ((I - A) * Inv)[i][j]; contraction over i in the einsum
  float q[8][8];
  for (int i = 0; i < 8; i++)
    for (int j = 0; j < 8; j++) {
      float s = 0.f;
      for (int m = 0; m < 8; m++)
        s += (((i == m) ? 1.f : 0.f) - A[i][m]) * Inv[m][j];
      q[i][j] = s;
    }

  // A-fragment table: Xt^T = Q^T @ X^T.  A[M=j][Kdim=i] = q[i][j].
  // chunk c covers input dims i = 4c..4c+3.
  float* aqp = aq + (size_t)tk * 128;
  for (int c = 0; c < 2; c++)
    for (int lane = 0; lane < 32; lane++) {
      const int h = lane >> 4, L = lane & 15;
      for (int r = 0; r < 2; r++)
        aqp[(c * 32 + lane) * 2 + r] = (L < 8) ? q[4 * c + 2 * h + r][L] : 0.f;
    }

  // rotated centers: ct[j] = sum_i q[i][j] * center[i]
  const float* cc = centers + tk * 8;
  for (int j = 0; j < 8; j++) {
    float s = 0.f;
    for (int i = 0; i < 8; i++) s += q[i][j] * cc[i];
    ct[tk * 8 + j] = s;
  }

  // base-2 per-dim variance terms:
  //   log2(gaussian) = lnc2 - xt^2 * ivl
  const float* lv = log_variance + tk * 8;
#pragma unroll
  for (int d = 0; d < 8; d++) {
    const float l   = fminf(fmaxf(lv[d], -MAXLV), MAXLV);
    const float var = expf(l);
    ivl[tk * 8 + d]  = (0.5f / var) * LOG2E_F;
    lnc2[tk * 8 + d] = -0.5f * (logf(TWO_PI_F * var) * LOG2E_F);
  }
}

// ---------------------------------------------------------------------------
// Kernel 2: clipped softmax over K=8 per t. 512 threads.
// ---------------------------------------------------------------------------
__global__ __launch_bounds__(256) void softmax_w_kernel(
    const float* __restrict__ wl, float* __restrict__ wout) {
  const int t = blockIdx.x * blockDim.x + threadIdx.x;
  if (t >= TDIM) return;
  float e[KDIM];
  float m = -1e30f;
#pragma unroll
  for (int k = 0; k < KDIM; k++) {
    e[k] = fminf(fmaxf(wl[t * KDIM + k], -GAINV), GAINV);
    m = fmaxf(m, e[k]);
  }
  float s = 0.f;
#pragma unroll
  for (int k = 0; k < KDIM; k++) { e[k] = expf(e[k] - m); s += e[k]; }
  const float r = 1.f / s;
#pragma unroll
  for (int k = 0; k < KDIM; k++) wout[t * KDIM + k] = e[k] * r;
}

// ---------------------------------------------------------------------------
// Kernel 3: main fused pass. One wave per (16-row b-tile, t).
//   Xt^T = Q^T @ x^T  via 2x V_WMMA_F32_16X16X4_F32, then xt -= ct (center
//   rotation folded out of the matmul by linearity).
// Fragment roles:
//   A = Q^T chunk (pre-swizzled table, unconditional b64 load per lane)
//   B = x^T       (N = 16 batch rows; k-loop invariant!)
//   C : lane n<16, VGPR j = xt_j of batch row b0+n  -> D-reduction is
//       entirely within-lane: no shuffles, no DS traffic.
// Per-dim mix in base-2 domain with raw hardware transcendentals:
//   g = exp2(u2), term = log2(g + EPS), p_k = exp2(sum term)
//   == prod_d (g_d + EPS)  -- identical math to the reference, but zero
//   base conversions and zero denorm-fixup code (log2 arg >= 1e-7).
// ---------------------------------------------------------------------------
__global__ __launch_bounds__(256) void mixture_pdf_wmma_kernel(
    const float* __restrict__ x,    // [B,T,8]
    const float* __restrict__ aq,   // [T,K,2,32,2]
    const float* __restrict__ ct,   // [T,K,8]
    const float* __restrict__ ivl,  // [T,K,8]
    const float* __restrict__ lnc2, // [T,K,8]
    const float* __restrict__ wbuf, // [T,K]
    float* __restrict__ out,        // [B,T]
    int B) {
  const int lane = threadIdx.x & 31;
  const int wid  = (int)((blockIdx.x * blockDim.x + threadIdx.x) >> 5);
  const int nbt  = B >> 4;  // # of 16-row b-tiles
  // wave-uniform by construction; force into SGPRs
  const int t  = __builtin_amdgcn_readfirstlane(wid / nbt);
  const int bt = __builtin_amdgcn_readfirstlane(wid - t * nbt);
  const int b0 = bt << 4;

  const int h = lane >> 4;   // K-pair selector within fragments
  const int n = lane & 15;   // B/C fragment column = batch row within tile

  // Load this lane's x row (8 contiguous floats, 16B aligned).
  const float* xp = x + (size_t)(b0 + n) * (TDIM * DDIM) + t * DDIM;
  float xr[8];
#pragma unroll
  for (int i = 0; i < 8; i++) xr[i] = xp[i];

  // B fragments (x^T): chunk c needs input dims {4c+2h, 4c+2h+1}.
  v2f bf0, bf1;
  bf0.x = h ? xr[2] : xr[0];
  bf0.y = h ? xr[3] : xr[1];
  bf1.x = h ? xr[6] : xr[4];
  bf1.y = h ? xr[7] : xr[5];

  const v2f* aqv = (const v2f*)aq;  // [T,K,2,32] of float2

  float acc = 0.f;
  for (int k = 0; k < KDIM; k++) {
    const int tk = t * KDIM + k;

    // A fragments: one coalesced b64 per chunk, no conditionals.
    const v2f af0 = aqv[(size_t)(tk * 2 + 0) * 32 + lane];
    const v2f af1 = aqv[(size_t)(tk * 2 + 1) * 32 + lane];

    v8f c = {};
    c = __builtin_amdgcn_wmma_f32_16x16x4_f32(false, af0, false, bf0,
                                              (short)0, c, false, false);
    c = __builtin_amdgcn_wmma_f32_16x16x4_f32(false, af1, false, bf1,
                                              (short)0, c, false, false);

    // wave-uniform per-(t,k) tables -> scalar loads
    const float wk = wbuf[tk];
    float lgpe2 = 0.f;
#pragma unroll
    for (int j = 0; j < 8; j++) {
      const float xt = c[j] - ct[tk * 8 + j];            // rotated, centered
      const float u2 = lnc2[tk * 8 + j] - xt * xt * ivl[tk * 8 + j];
      const float g  = __builtin_amdgcn_exp2f(u2);       // gaussian value
      lgpe2 += __builtin_amdgcn_logf(g + EPSV);          // log2(g + EPS)
    }
    acc += __builtin_amdgcn_exp2f(lgpe2) * wk;           // prod(g_d+EPS)*w_k
  }

  // lanes 0..15 hold p for rows b0..b0+15 (lanes 16..31 computed pad dims)
  if (lane < 16) out[(size_t)(b0 + lane) * TDIM + t] = acc;
}

// ---------------------------------------------------------------------------
extern "C" void kernel_launch(void* const* d_in, const int* in_sizes, int n_in,
                              void* d_out, int out_size, void* d_ws, size_t ws_size,
                              hipStream_t stream) {
  const float* x       = (const float*)d_in[0];  // [B,T,8]
  const float* centers = (const float*)d_in[1];  // [T,K,8]
  const float* wl      = (const float*)d_in[2];  // [T,K]
  const float* lv      = (const float*)d_in[3];  // [T,K,8]
  const float* cov     = (const float*)d_in[4];  // [T,K,28]
  float* out = (float*)d_out;

  const int B = in_sizes[0] / (TDIM * DDIM);

  // workspace carve-up (floats)
  float* ws   = (float*)d_ws;
  float* aq   = ws;                          // T*K*128
  float* ct   = aq + TDIM * KDIM * 128;      // T*K*8
  float* ivl  = ct + TDIM * KDIM * 8;        // T*K*8
  float* lnc2 = ivl + TDIM * KDIM * 8;       // T*K*8
  float* wbuf = lnc2 + TDIM * KDIM * 8;      // T*K

  // 1) Cayley transforms + fragment tables: 4096 threads
  cayley_setup_kernel<<<(TDIM * KDIM + 255) / 256, 256, 0, stream>>>(
      cov, lv, centers, aq, ct, ivl, lnc2);

  // 2) softmax weights: 512 threads
  softmax_w_kernel<<<(TDIM + 255) / 256, 256, 0, stream>>>(wl, wbuf);

  // 3) main fused pass: one wave per (16-row tile, t); exact grid, no guards
  const int nbt    = B >> 4;                 // 128
  const long waves = (long)nbt * TDIM;       // 65536
  const long thr   = waves * 32;
  mixture_pdf_wmma_kernel<<<(int)(thr / 256), 256, 0, stream>>>(
      x, aq, ct, ivl, lnc2, wbuf, out, B);
}